// GCN3D_86998857548340
// MI455X (gfx1250) — compile-verified
//
#include <hip/hip_runtime.h>

typedef __attribute__((ext_vector_type(2))) float v2f;
typedef __attribute__((ext_vector_type(8))) float v8f;

#define NNODES 50000
#define NEDGES 800000
#define FIN 64
#define HID 128
#define NCLS 2
#define NGR 16

// ---------------- degree / norm ----------------
__global__ void deg_init_kernel(float* deg) {
    int n = blockIdx.x * blockDim.x + threadIdx.x;
    if (n < NNODES) deg[n] = 1.0f;  // self loop
}

__global__ void deg_count_kernel(const int* ei, float* deg) {
    int e = blockIdx.x * blockDim.x + threadIdx.x;
    if (e < NEDGES) atomicAdd(&deg[ei[NEDGES + e]], 1.0f);
}

__global__ void dinv_kernel(float* deg) {
    int n = blockIdx.x * blockDim.x + threadIdx.x;
    if (n < NNODES) deg[n] = 1.0f / sqrtf(deg[n]);  // deg >= 1 always
}

// ---------------- WMMA fp32 matmul + fused agg init ------------------------
// Out[N x 128] = A[N x K] @ W[K x 128]        (hw, consumed by edge gather)
// Agg[N x 128] = bias + Out * dinv^2          (self-loop message + bias, pre-scatter)
// block: (32, 8) = 8 waves; each wave owns one 16-col tile; one 16-row tile per block.
// relu_in: apply ReLU to A while staging into LDS (fuses previous layer's activation).
__global__ void __launch_bounds__(256)
matmul_wmma_kernel(const float* __restrict__ A, const float* __restrict__ W,
                   float* __restrict__ Out, float* __restrict__ Agg,
                   const float* __restrict__ dinv, const float* __restrict__ bias,
                   int K, int relu_in) {
    __shared__ float As[16 * (128 + 4)];
    __shared__ float sd2[16];
    const int ldA = K + 4;
    const int tid = threadIdx.y * 32 + threadIdx.x;
    const int row0 = blockIdx.x * 16;

    if (tid < 16) {
        float d = dinv[row0 + tid];
        sd2[tid] = d * d;
    }
    // cooperative load of 16 x K tile (float4), fused ReLU
    const int nvec = 16 * (K >> 2);
    for (int i = tid; i < nvec; i += 256) {
        int r = i / (K >> 2);
        int c4 = (i - r * (K >> 2)) << 2;
        float4 v = *(const float4*)(A + (long)(row0 + r) * K + c4);
        if (relu_in) {
            v.x = fmaxf(v.x, 0.0f); v.y = fmaxf(v.y, 0.0f);
            v.z = fmaxf(v.z, 0.0f); v.w = fmaxf(v.w, 0.0f);
        }
        *(float4*)(&As[r * ldA + c4]) = v;
    }
    __syncthreads();

    const int lane = threadIdx.x;
    const int half = lane >> 4;     // 0: lanes 0-15, 1: lanes 16-31
    const int m    = lane & 15;
    const int col0 = threadIdx.y * 16;

    v8f c = {};
    for (int k0 = 0; k0 < K; k0 += 4) {
        int ka = k0 + half * 2;
        v2f a, b;
        a.x = As[m * ldA + ka];
        a.y = As[m * ldA + ka + 1];
        b.x = W[(long)ka * HID + col0 + m];
        b.y = W[(long)(ka + 1) * HID + col0 + m];
        c = __builtin_amdgcn_wmma_f32_16x16x4_f32(
                /*neg_a=*/false, a, /*neg_b=*/false, b,
                /*c_mod=*/(short)0, c, /*reuse_a=*/false, /*reuse_b=*/false);
    }

    // C layout: VGPR v -> M = v + half*8, N = lane&15
    const float bv = bias[col0 + m];
    for (int v = 0; v < 8; ++v) {
        int row = row0 + v + half * 8;
        float val = c[v];
        long o = (long)row * HID + col0 + m;
        Out[o] = val;
        Agg[o] = bv + val * sd2[v + half * 8];
    }
}

// ---------------- edge scatter: one wave per edge, float4 per lane ---------
__global__ void __launch_bounds__(256)
edge_scatter_kernel(const int* __restrict__ ei, const float* __restrict__ dinv,
                    const float* __restrict__ hw, float* __restrict__ agg) {
    int e = blockIdx.x * 8 + threadIdx.y;
    if (e >= NEDGES) return;
    int src = ei[e];
    int dst = ei[NEDGES + e];
    float nrm = dinv[src] * dinv[dst];
    int f4 = threadIdx.x * 4;
    float4 v = *(const float4*)(hw + (long)src * HID + f4);
    float* a = agg + (long)dst * HID + f4;
    atomicAdd(a + 0, v.x * nrm);
    atomicAdd(a + 1, v.y * nrm);
    atomicAdd(a + 2, v.z * nrm);
    atomicAdd(a + 3, v.w * nrm);
}

// ---------------- pooling -------------------------------------------------
__global__ void pool_zero_kernel(float* pooled, float* cnt) {
    int i = blockIdx.x * blockDim.x + threadIdx.x;
    if (i < NGR * HID) pooled[i] = 0.0f;
    if (i < NGR) cnt[i] = 0.0f;
}

__global__ void __launch_bounds__(256)
pool_kernel(const float* __restrict__ h, const int* __restrict__ batch,
            float* __restrict__ pooled, float* __restrict__ cnt) {
    __shared__ float sp[NGR * HID];
    __shared__ float sc[NGR];
    int tid = threadIdx.x;
    for (int i = tid; i < NGR * HID; i += 256) sp[i] = 0.0f;
    if (tid < NGR) sc[tid] = 0.0f;
    __syncthreads();
    const long total = (long)NNODES * HID;
    for (long idx = (long)blockIdx.x * 256 + tid; idx < total; idx += (long)gridDim.x * 256) {
        int n = (int)(idx >> 7), f = (int)(idx & 127);
        int g = batch[n];
        atomicAdd(&sp[g * HID + f], h[idx]);
        if (f == 0) atomicAdd(&sc[g], 1.0f);
    }
    __syncthreads();
    for (int i = tid; i < NGR * HID; i += 256) atomicAdd(&pooled[i], sp[i]);
    if (tid < NGR) atomicAdd(&cnt[tid], sc[tid]);
}

// ---------------- head MLP (single block) ---------------------------------
__global__ void __launch_bounds__(256)
head_kernel(const float* __restrict__ pooled, const float* __restrict__ cnt,
            const float* __restrict__ lw1, const float* __restrict__ lb1,
            const float* __restrict__ lw2, const float* __restrict__ lb2,
            float* __restrict__ out) {
    __shared__ float p[NGR * HID];
    __shared__ float hid[NGR * 64];
    int tid = threadIdx.x;
    for (int i = tid; i < NGR * HID; i += 256) {
        int g = i >> 7;
        p[i] = pooled[i] / fmaxf(cnt[g], 1.0f);
    }
    __syncthreads();
    for (int i = tid; i < NGR * 64; i += 256) {
        int g = i >> 6, j = i & 63;
        float s = lb1[j];
        for (int k = 0; k < HID; ++k) s += p[g * HID + k] * lw1[k * 64 + j];
        hid[i] = fmaxf(s, 0.0f);
    }
    __syncthreads();
    if (tid < NGR * NCLS) {
        int g = tid >> 1, c0 = tid & 1;
        float s = lb2[c0];
        for (int k = 0; k < 64; ++k) s += hid[g * 64 + k] * lw2[k * NCLS + c0];
        out[tid] = s;
    }
}

// ---------------- launch --------------------------------------------------
extern "C" void kernel_launch(void* const* d_in, const int* in_sizes, int n_in,
                              void* d_out, int out_size, void* d_ws, size_t ws_size,
                              hipStream_t stream) {
    const float* x  = (const float*)d_in[0];
    const int*   ei = (const int*)d_in[1];
    const int*   batch = (const int*)d_in[2];
    const float* W1 = (const float*)d_in[3];  const float* b1 = (const float*)d_in[4];
    const float* W2 = (const float*)d_in[5];  const float* b2 = (const float*)d_in[6];
    const float* W3 = (const float*)d_in[7];  const float* b3 = (const float*)d_in[8];
    const float* W4 = (const float*)d_in[9];  const float* b4 = (const float*)d_in[10];
    const float* lw1 = (const float*)d_in[11]; const float* lb1 = (const float*)d_in[12];
    const float* lw2 = (const float*)d_in[13]; const float* lb2 = (const float*)d_in[14];
    float* out = (float*)d_out;

    char* ws = (char*)d_ws;
    size_t off = 0;
    float* dinv = (float*)(ws + off); off += ((size_t)NNODES * 4 + 255) & ~(size_t)255;
    float* buf0 = (float*)(ws + off); off += (size_t)NNODES * HID * 4;
    float* buf1 = (float*)(ws + off); off += (size_t)NNODES * HID * 4;
    float* pooled = (float*)(ws + off); off += NGR * HID * 4;
    float* cnt = (float*)(ws + off); off += NGR * 4;
    (void)ws_size; (void)in_sizes; (void)n_in; (void)out_size;

    // norms
    deg_init_kernel<<<(NNODES + 255) / 256, 256, 0, stream>>>(dinv);
    deg_count_kernel<<<(NEDGES + 255) / 256, 256, 0, stream>>>(ei, dinv);
    dinv_kernel<<<(NNODES + 255) / 256, 256, 0, stream>>>(dinv);

    dim3 mmBlock(32, 8);
    dim3 edBlock(32, 8);
    const int mmGrid = NNODES / 16;          // 3125, exact
    const int edGrid = (NEDGES + 7) / 8;     // 100000

    // layer 1: x[50000x64] @ W1 -> hw in buf1, agg(bias+selfloop) in buf0
    matmul_wmma_kernel<<<mmGrid, mmBlock, 0, stream>>>(x, W1, buf1, buf0, dinv, b1, FIN, 0);
    edge_scatter_kernel<<<edGrid, edBlock, 0, stream>>>(ei, dinv, buf1, buf0);

    // layer 2 (ReLU of layer-1 output fused into A load)
    matmul_wmma_kernel<<<mmGrid, mmBlock, 0, stream>>>(buf0, W2, buf1, buf0, dinv, b2, HID, 1);
    edge_scatter_kernel<<<edGrid, edBlock, 0, stream>>>(ei, dinv, buf1, buf0);

    // layer 3
    matmul_wmma_kernel<<<mmGrid, mmBlock, 0, stream>>>(buf0, W3, buf1, buf0, dinv, b3, HID, 1);
    edge_scatter_kernel<<<edGrid, edBlock, 0, stream>>>(ei, dinv, buf1, buf0);

    // layer 4 (no ReLU after)
    matmul_wmma_kernel<<<mmGrid, mmBlock, 0, stream>>>(buf0, W4, buf1, buf0, dinv, b4, HID, 1);
    edge_scatter_kernel<<<edGrid, edBlock, 0, stream>>>(ei, dinv, buf1, buf0);

    // mean pool + head
    pool_zero_kernel<<<(NGR * HID + 255) / 256, 256, 0, stream>>>(pooled, cnt);
    pool_kernel<<<120, 256, 0, stream>>>(buf0, batch, pooled, cnt);
    head_kernel<<<1, 256, 0, stream>>>(pooled, cnt, lw1, lb1, lw2, lb2, out);
}